// SimplicialConvolution2_43198781063247
// MI455X (gfx1250) — compile-verified
//
#include <hip/hip_runtime.h>

typedef float v2f __attribute__((ext_vector_type(2)));
typedef float v8f __attribute__((ext_vector_type(8)));

constexpr int MM   = 4096;   // graph size M
constexpr int CIN  = 256;
constexpr int COUT = 256;
constexpr int BB   = 8;      // batch
constexpr int KTH  = 8;      // K1+K2 (theta last dim)

constexpr int BR    = 64;    // output rows per block
constexpr int BN    = 256;   // output cols per block
constexpr int KC    = 32;    // K chunk staged in LDS
constexpr int PITCH = KC + 4; // 36 floats: 16B-aligned rows, bank-conflict-free

// MODE 0:  D += V * L^T      (V per-batch (COUT x MM), L (MM x MM) row-major)
// MODE 1:  D  = ThetaSum * x (+ bias)   channel mix, contraction over CIN
template <int MODE>
__global__ __launch_bounds__(256)
void cheb_wmma_kernel(const float* __restrict__ Aop,
                      const float* __restrict__ Bop,
                      const float* __restrict__ bias,
                      float* __restrict__ D,
                      int ka, int kb)
{
  __shared__ float ldsA[BR * PITCH];
  __shared__ float ldsB[BN * PITCH];

  const int tid  = threadIdx.x;
  const int lane = tid & 31;
  const int wave = tid >> 5;
  const int wr   = wave & 1;    // 0..1 : 32-row group
  const int wc   = wave >> 1;   // 0..3 : 64-col group
  const int half = lane >> 4;   // lane 16..31 -> K/M upper half
  const int l16  = lane & 15;

  const int nBase = blockIdx.x * BN;
  const int rBase = blockIdx.y * BR;
  const int z     = blockIdx.z;
  const size_t zOut = (size_t)z * COUT * MM;

  constexpr int KTOT = (MODE == 0) ? MM : CIN;

  v8f acc[2][4];
  if constexpr (MODE == 0) {
    // accumulate semantics: seed accumulators with current D (C/D VGPR layout)
    #pragma unroll
    for (int t = 0; t < 2; ++t)
      #pragma unroll
      for (int s = 0; s < 4; ++s)
        #pragma unroll
        for (int v = 0; v < 8; ++v) {
          const int r = rBase + wr*32 + t*16 + v + 8*half;
          const int n = nBase + wc*64 + s*16 + l16;
          acc[t][s][v] = D[zOut + (size_t)r*MM + n];
        }
  } else {
    const v8f zero = {};
    #pragma unroll
    for (int t = 0; t < 2; ++t)
      #pragma unroll
      for (int s = 0; s < 4; ++s)
        acc[t][s] = zero;
  }

  for (int k0 = 0; k0 < KTOT; k0 += KC) {
    __syncthreads();
    if constexpr (MODE == 0) {
      const int row  = tid >> 3;  // 0..31
      const int quad = tid & 7;   // float4 slot along K
      // A tile: V rows (K-contiguous) -> ldsA[r][k]
      #pragma unroll
      for (int p = 0; p < 2; ++p) {
        const int r = row + 32*p;
        const float4 v4 = *(const float4*)(Aop + zOut + (size_t)(rBase + r)*MM + k0 + quad*4);
        *(float4*)&ldsA[r*PITCH + quad*4] = v4;
      }
      // B tile: L rows (K-contiguous, since R = V*L^T) -> ldsB[n][k]
      #pragma unroll
      for (int p = 0; p < 8; ++p) {
        const int n = row + 32*p;
        const float* src = Bop + (size_t)(nBase + n)*MM + k0 + quad*4;
        *(float4*)&ldsB[n*PITCH + quad*4] = *(const float4*)src;
        if (k0 + KC < KTOT) __builtin_prefetch(src + KC, 0, 0);  // global_prefetch_b8
      }
    } else {
      // A tile: ThetaSum[o][i] = theta[o][i][ka] (+ theta[o][i][kb]); stride-8 gather
      const int col  = tid & 31;  // i within chunk
      const int rowa = tid >> 5;  // 0..7
      #pragma unroll
      for (int p = 0; p < 8; ++p) {
        const int o = rowa + 8*p;
        const float* tsrc = Aop + (size_t)(rBase + o)*(CIN*KTH) + (size_t)(k0 + col)*KTH;
        float val = tsrc[ka];
        if (kb >= 0) val += tsrc[kb];
        ldsA[o*PITCH + col] = val;
      }
      // B tile: x[z][i][m] is m-contiguous; transpose-stage into ldsB[m][i]
      const int kq = tid >> 6;    // 0..3
      const int nq = tid & 63;    // float4 slot along m
      const size_t zIn = (size_t)z * CIN * MM;
      #pragma unroll
      for (int p = 0; p < 8; ++p) {
        const int i = kq + 4*p;
        const float4 v4 = *(const float4*)(Bop + zIn + (size_t)(k0 + i)*MM + nBase + nq*4);
        ldsB[(nq*4 + 0)*PITCH + i] = v4.x;
        ldsB[(nq*4 + 1)*PITCH + i] = v4.y;
        ldsB[(nq*4 + 2)*PITCH + i] = v4.z;
        ldsB[(nq*4 + 3)*PITCH + i] = v4.w;
      }
    }
    __syncthreads();

    #pragma unroll
    for (int kk = 0; kk < KC; kk += 4) {
      // A frag: element [l16][2*half + v] ; B frag: element [2*half + v][l16]
      v2f a[2], b[4];
      #pragma unroll
      for (int t = 0; t < 2; ++t)
        a[t] = *(const v2f*)&ldsA[(wr*32 + t*16 + l16)*PITCH + kk + 2*half];
      #pragma unroll
      for (int s = 0; s < 4; ++s)
        b[s] = *(const v2f*)&ldsB[(wc*64 + s*16 + l16)*PITCH + kk + 2*half];
      #pragma unroll
      for (int t = 0; t < 2; ++t)
        #pragma unroll
        for (int s = 0; s < 4; ++s)
          acc[t][s] = __builtin_amdgcn_wmma_f32_16x16x4_f32(
              false, a[t], false, b[s], (short)0, acc[t][s], false, false);
    }
  }

  #pragma unroll
  for (int t = 0; t < 2; ++t)
    #pragma unroll
    for (int s = 0; s < 4; ++s)
      #pragma unroll
      for (int v = 0; v < 8; ++v) {
        const int r = rBase + wr*32 + t*16 + v + 8*half;
        const int n = nBase + wc*64 + s*16 + l16;
        float val = acc[t][s][v];
        if constexpr (MODE == 1) {
          if (bias) val += bias[r];   // bias1 is (1, COUT, 1)
        }
        D[zOut + (size_t)r*MM + n] = val;
      }
}

// y = u0 + u4 + bias + Ll*(u1 + Ll*(u2 + Ll*u3)) + Lu*(u5 + Lu*(u6 + Lu*u7))
// where u_k = Theta[:,:,k] applied over channels. Needs 2 x 32MiB workspace.
extern "C" void kernel_launch(void* const* d_in, const int* in_sizes, int n_in,
                              void* d_out, int out_size, void* d_ws, size_t ws_size,
                              hipStream_t stream) {
  (void)in_sizes; (void)n_in; (void)out_size; (void)ws_size;
  const float* Ll   = (const float*)d_in[0];
  const float* Lu   = (const float*)d_in[1];
  const float* x    = (const float*)d_in[2];
  const float* th   = (const float*)d_in[3];
  const float* bias = (const float*)d_in[4];
  float* out = (float*)d_out;
  float* w0 = (float*)d_ws;
  float* w1 = w0 + (size_t)BB * COUT * MM;

  dim3 grid(MM / BN, COUT / BR, BB);
  dim3 block(256);

  auto MIX = [&](float* dst, int ka, int kb, const float* b) {
    cheb_wmma_kernel<1><<<grid, block, 0, stream>>>(th, x, b, dst, ka, kb);
  };
  auto ACC = [&](const float* V, const float* L, float* dst) {
    cheb_wmma_kernel<0><<<grid, block, 0, stream>>>(V, L, nullptr, dst, 0, 0);
  };

  // Ll Horner chain
  MIX(w0, 3, -1, nullptr);        // w0 = u3
  MIX(w1, 2, -1, nullptr);        // w1 = u2
  ACC(w0, Ll, w1);                // w1 += Ll*w0
  MIX(w0, 1, -1, nullptr);        // w0 = u1
  ACC(w1, Ll, w0);                // w0 += Ll*w1
  MIX(out, 0, 4, bias);           // out = u0 + u4 + bias
  ACC(w0, Ll, out);               // out += Ll*w0

  // Lu Horner chain
  MIX(w0, 7, -1, nullptr);        // w0 = u7
  MIX(w1, 6, -1, nullptr);        // w1 = u6
  ACC(w0, Lu, w1);                // w1 += Lu*w0
  MIX(w0, 5, -1, nullptr);        // w0 = u5
  ACC(w1, Lu, w0);                // w0 += Lu*w1
  ACC(w0, Lu, out);               // out += Lu*w0
}